// QuantizedLinear_36679020708432
// MI455X (gfx1250) — compile-verified
//
#include <hip/hip_runtime.h>
#include <hip/hip_bf16.h>

// Problem constants (match reference)
#define IN_FEATURES   8192
#define OUT_FEATURES  8192
#define GRP           8        // in_group_size
#define NCB           2        // num codebooks
#define KSZ           256      // codebook size
#define NJ            1024     // num in groups = IN_FEATURES / GRP
#define CKTOT         (NCB * KSZ)   // 512 lut columns per j
#define J_SPLIT       8
#define J_PER_SPLIT   (NJ / J_SPLIT)   // 128
#define J_CHUNK       16               // LDS chunk: 16*512*4 = 32KB (x2 buffers)
#define NCHUNK        (J_PER_SPLIT / J_CHUNK)  // 8
#define O_TILE        256

typedef __attribute__((ext_vector_type(2))) float v2f;
typedef __attribute__((ext_vector_type(8))) float v8f;

// Operand types for the async global->LDS b128 builtin (typed pointers on
// this toolchain: vector_size(16) int, AS(1) source / AS(3) destination).
typedef int v4i __attribute__((vector_size(16)));
typedef __attribute__((address_space(1))) v4i gv4i;
typedef __attribute__((address_space(3))) v4i lv4i;

#if __has_builtin(__builtin_amdgcn_global_load_async_to_lds_b128) && \
    __has_builtin(__builtin_amdgcn_s_wait_asynccnt)
#define HAS_ASYNC_LDS 1
#else
#define HAS_ASYNC_LDS 0
#endif

// ---------------------------------------------------------------------------
// Kernel 1: LUT build via f32 WMMA.
// D(1024x512) = A(1024x8) * B(8x512), A[j][t] = x[j*8+t], B[t][ck] = codebooks[ck*8+t].
// One wave32 per 16x16 output tile; K=8 handled as two chained 16x16x4 WMMAs.
// ---------------------------------------------------------------------------
__global__ __launch_bounds__(256) void lut_wmma_kernel(
    const float* __restrict__ x,          // (8192,)
    const float* __restrict__ codebooks,  // (2,256,1,8) flat -> (512,8)
    float* __restrict__ lut)              // (1024,512)
{
    const int wave = (blockIdx.x * blockDim.x + threadIdx.x) >> 5;
    const int lane = threadIdx.x & 31;
    const int jt  = wave >> 5;   // 0..63  (j tile)
    const int ckt = wave & 31;   // 0..31  (ck tile)

    const int m  = lane & 15;    // row within tile (A) / col within tile (B,D)
    const int kh = lane >> 4;    // k-half select

    // A fragment: 16x4 f32 layout -> VGPR0 = A[m][2*kh], VGPR1 = A[m][2*kh+1]
    const float* xr = x + (jt * 16 + m) * GRP;
    v2f a0; a0[0] = xr[2 * kh];     a0[1] = xr[2 * kh + 1];
    v2f a1; a1[0] = xr[4 + 2 * kh]; a1[1] = xr[4 + 2 * kh + 1];

    // B fragment: 4x16 f32, symmetric striping -> VGPR0 = B[2*kh][n], VGPR1 = B[2*kh+1][n]
    const float* cbr = codebooks + (ckt * 16 + m) * GRP;
    v2f b0; b0[0] = cbr[2 * kh];     b0[1] = cbr[2 * kh + 1];
    v2f b1; b1[0] = cbr[4 + 2 * kh]; b1[1] = cbr[4 + 2 * kh + 1];

    v8f acc = {};
    acc = __builtin_amdgcn_wmma_f32_16x16x4_f32(false, a0, false, b0,
                                                (short)0, acc, false, false);
    acc = __builtin_amdgcn_wmma_f32_16x16x4_f32(false, a1, false, b1,
                                                (short)0, acc, false, false);

    // D: VGPR r holds D[M = r + 8*kh][n = m]
    const int colBase = ckt * 16 + m;
    #pragma unroll
    for (int r = 0; r < 8; ++r) {
        lut[(jt * 16 + r + 8 * kh) * CKTOT + colBase] = acc[r];
    }
}

// ---------------------------------------------------------------------------
// Kernel 2: bandwidth-bound gather. Each block: one 256-output tile x one
// 128-group j-slice. LUT slices double-buffered in LDS via async global->LDS
// (ASYNCcnt) so staging overlaps the coalesced int2 codes stream.
// Deterministic partials (no float atomics).
// ---------------------------------------------------------------------------
__global__ __launch_bounds__(256) void gather_kernel(
    const int2* __restrict__ codes2,   // (1024, 8192) int2 (code pair per (j,o))
    const float* __restrict__ lut,     // (1024, 512)
    float* __restrict__ partial)       // (J_SPLIT, 8192)
{
    __shared__ __align__(16) float slut[2][J_CHUNK * CKTOT];   // 2 x 32 KB

    const int tid = threadIdx.x;
    const int o   = blockIdx.x * O_TILE + tid;
    const int j0  = blockIdx.y * J_PER_SPLIT;

    float acc = 0.0f;

#if HAS_ASYNC_LDS
    // Each thread owns 8 b128 transfers per 32KB chunk (2048 float4 / 256 thr).
    auto issue_chunk = [&](int buf, int chunk) {
        const float4* src = (const float4*)(lut + (size_t)(j0 + chunk * J_CHUNK) * CKTOT);
        float4* dst = (float4*)slut[buf];
        #pragma unroll
        for (int i = 0; i < (J_CHUNK * CKTOT / 4) / O_TILE; ++i) {
            __builtin_amdgcn_global_load_async_to_lds_b128(
                (gv4i*)(src + i * O_TILE + tid),
                (lv4i*)(dst + i * O_TILE + tid),
                0, 0);
        }
    };

    issue_chunk(0, 0);
    for (int c = 0; c < NCHUNK; ++c) {
        const int buf = c & 1;
        __builtin_amdgcn_s_wait_asynccnt(0);   // my wave's chunk-c loads landed
        __syncthreads();                       // all waves landed; prev buffer free
        if (c + 1 < NCHUNK) issue_chunk(buf ^ 1, c + 1);

        const float* cl = slut[buf];
        #pragma unroll 4
        for (int jj = 0; jj < J_CHUNK; ++jj) {
            const int j = j0 + c * J_CHUNK + jj;
            int2 cd = codes2[(size_t)j * OUT_FEATURES + o];
            acc += cl[jj * CKTOT + cd.x] + cl[jj * CKTOT + KSZ + cd.y];
        }
    }
#else
    // Fallback: synchronous staging.
    for (int c = 0; c < NCHUNK; ++c) {
        const float4* src = (const float4*)(lut + (size_t)(j0 + c * J_CHUNK) * CKTOT);
        float4* dst = (float4*)slut[0];
        #pragma unroll
        for (int i = 0; i < (J_CHUNK * CKTOT / 4) / O_TILE; ++i) {
            dst[i * O_TILE + tid] = src[i * O_TILE + tid];
        }
        __syncthreads();
        #pragma unroll 4
        for (int jj = 0; jj < J_CHUNK; ++jj) {
            const int j = j0 + c * J_CHUNK + jj;
            int2 cd = codes2[(size_t)j * OUT_FEATURES + o];
            acc += slut[0][jj * CKTOT + cd.x] + slut[0][jj * CKTOT + KSZ + cd.y];
        }
        __syncthreads();
    }
#endif

    partial[blockIdx.y * OUT_FEATURES + o] = acc;
}

// ---------------------------------------------------------------------------
// Kernel 3: reduce partials, apply scale + bias.
// ---------------------------------------------------------------------------
__global__ __launch_bounds__(256) void reduce_kernel(
    const float* __restrict__ partial,
    const float* __restrict__ scales,
    const float* __restrict__ bias,
    float* __restrict__ out)
{
    const int o = blockIdx.x * blockDim.x + threadIdx.x;
    float s = 0.0f;
    #pragma unroll
    for (int p = 0; p < J_SPLIT; ++p) s += partial[p * OUT_FEATURES + o];
    out[o] = s * scales[o] + bias[o];
}

extern "C" void kernel_launch(void* const* d_in, const int* in_sizes, int n_in,
                              void* d_out, int out_size, void* d_ws, size_t ws_size,
                              hipStream_t stream) {
    const float* x         = (const float*)d_in[0];
    const float* codebooks = (const float*)d_in[1];
    const float* scales    = (const float*)d_in[2];
    const float* bias      = (const float*)d_in[3];
    const int*   codes     = (const int*)d_in[4];
    float*       out       = (float*)d_out;

    float* lut     = (float*)d_ws;                 // 1024*512 floats = 2 MiB
    float* partial = lut + (size_t)NJ * CKTOT;     // J_SPLIT*8192 floats = 256 KiB

    // 2048 tile-waves, 8 waves per block -> 256 blocks
    lut_wmma_kernel<<<256, 256, 0, stream>>>(x, codebooks, lut);

    gather_kernel<<<dim3(OUT_FEATURES / O_TILE, J_SPLIT), O_TILE, 0, stream>>>(
        (const int2*)codes, lut, partial);

    reduce_kernel<<<OUT_FEATURES / 256, 256, 0, stream>>>(partial, scales, bias, out);
}